// Sampler_59691455480289
// MI455X (gfx1250) — compile-verified
//
#include <hip/hip_runtime.h>
#include <stdint.h>

#define ASG __attribute__((address_space(1)))
#define ASL __attribute__((address_space(3)))

typedef int v4i __attribute__((vector_size(16)));

static constexpr int   B     = 256;
static constexpr int   V     = 128000;
static constexpr int   L     = 256;
static constexpr int   NLP   = 20;      // num_logprobs
static constexpr int   BLK   = 256;     // 8 waves (wave32)
static constexpr int   TILE  = 1024;    // floats per LDS tile (4 KB) -> 4 floats/thread (b128)
static constexpr int   NT    = V / TILE; // 125, exact
static constexpr int   PERT  = 8;       // per-thread candidates
static constexpr int   CAND  = BLK * PERT; // 2048 candidates per row
static constexpr int   NSORT = 64;      // sorted processed values kept (>= max top_k=49)
static constexpr int   HSZ   = 1024;    // LDS hash slots
static constexpr float FEPS  = 1e-5f;

// workspace layout (floats) per row:
//  [0]=M_raw [1]=S_raw [2]=M_proc [3]=S_proc [4]=final_threshold  (rest pad)
static constexpr int HDR  = 16;
static constexpr int ROWF = HDR + 4 * CAND; // 8208 floats/row  (~8.0 MiB total)

#define NEGINF (-__builtin_inff())

// ---------------- CDNA5 async global->LDS copy (ASYNCcnt path) ----------------
__device__ __forceinline__ void async_copy_b128(const float* g, float* l) {
#if defined(__AMDGCN__) && __has_builtin(__builtin_amdgcn_global_load_async_to_lds_b128)
  __builtin_amdgcn_global_load_async_to_lds_b128(
      (ASG v4i*)(uintptr_t)g, (ASL v4i*)(ASL void*)l, 0, 0);
#else
  ASL float* lp = (ASL float*)(ASL void*)l;
  asm volatile("global_load_async_to_lds_b128 %0, %1, off"
               :: "v"(lp), "v"(g) : "memory");
#endif
}
__device__ __forceinline__ void wait_async_le1() {
#if defined(__AMDGCN__) && __has_builtin(__builtin_amdgcn_s_wait_asynccnt)
  __builtin_amdgcn_s_wait_asynccnt(1);
#else
  asm volatile("s_wait_asynccnt 0x1" ::: "memory");
#endif
}
__device__ __forceinline__ void wait_async_0() {
#if defined(__AMDGCN__) && __has_builtin(__builtin_amdgcn_s_wait_asynccnt)
  __builtin_amdgcn_s_wait_asynccnt(0);
#else
  asm volatile("s_wait_asynccnt 0x0" ::: "memory");
#endif
}

// ---------------- helpers ----------------
__device__ __forceinline__ int hlookup(const int* hkey, const int* hcnt, int v) {
  unsigned h = ((unsigned)v * 2654435761u) & (HSZ - 1u);
#pragma unroll 1
  for (;;) {
    int k = hkey[h];
    if (k == v) return hcnt[h];
    if (k == -1) return 0;
    h = (h + 1) & (HSZ - 1u);
  }
}

__device__ __forceinline__ void build_hash(int* hkey, int* hcnt, const int* ids_row,
                                           int tid) {
  for (int i = tid; i < HSZ; i += BLK) { hkey[i] = -1; hcnt[i] = 0; }
  __syncthreads();
  for (int i = tid; i < L; i += BLK) {
    int id = ids_row[i];
    unsigned h = ((unsigned)id * 2654435761u) & (HSZ - 1u);
#pragma unroll 1
    for (;;) {
      int prev = atomicCAS(&hkey[h], -1, id);
      if (prev == -1 || prev == id) { atomicAdd(&hcnt[h], 1); break; }
      h = (h + 1) & (HSZ - 1u);
    }
  }
  __syncthreads();
}

__device__ __forceinline__ float proc_val(float x, int cnt, float rp, float fp,
                                          float pp, float invt) {
  float y = x;
  if (cnt > 0) {
    y = (x > 0.0f) ? (x / rp) : (x * rp);
    y = y - (float)cnt * fp - pp;
  }
  return y * invt;
}

// branchy online logsumexp update: ~1 exp per call after warm-up
__device__ __forceinline__ void lse_update(float& M, float& S, float x) {
  if (x > M) {
    S = S * __expf(M - x) + 1.0f; // exp(-inf)=0 handles first element
    M = x;
  } else {
    S += __expf(x - M);
  }
}

__device__ __forceinline__ void topk_insert(float* vals, int* idxs, float x, int id) {
  if (x <= vals[PERT - 1]) return;
  int p = PERT - 1;
#pragma unroll 1
  while (p > 0 && x > vals[p - 1]) {
    vals[p] = vals[p - 1]; idxs[p] = idxs[p - 1]; --p;
  }
  vals[p] = x; idxs[p] = id;
}

__device__ __forceinline__ float rng_uniform(uint64_t k) {
  k ^= 42ull * 0x9E3779B97F4A7C15ull;
  k ^= k >> 33; k *= 0xff51afd7ed558ccdULL;
  k ^= k >> 33; k *= 0xc4ceb9fe1a85ec53ULL;
  k ^= k >> 33;
  unsigned r = (unsigned)(k >> 32);
  return ((float)r + 0.5f) * 2.3283064365386963e-10f; // (0,1)
}

// =====================================================================
// Kernel 1: streamed pass -> LSE stats (raw [+ processed if needed]) + candidates
// =====================================================================
__global__ void __launch_bounds__(BLK) k1_stats(
    const float* __restrict__ logits, const float* __restrict__ temperature,
    const int* __restrict__ top_k, const float* __restrict__ rep_pen,
    const float* __restrict__ freq_pen, const float* __restrict__ pres_pen,
    const int* __restrict__ out_ids, float* __restrict__ ws) {
  __shared__ float buf[2][TILE];
  __shared__ int   hkey[HSZ];
  __shared__ int   hcnt[HSZ];
  __shared__ float rm[BLK], rs[BLK], pm[BLK], ps[BLK];

  const int b   = blockIdx.x;
  const int tid = threadIdx.x;
  const float* row = logits + (size_t)b * V;

  build_hash(hkey, hcnt, out_ids + (size_t)b * L, tid);

  const float rp = rep_pen[b], fp = freq_pen[b], pp = pres_pen[b];
  float t = temperature[b];
  t = (t < FEPS) ? 1.0f : t;
  const float invt = 1.0f / t;
  const bool needProcLse = (top_k[b] <= 0); // proc denominator only used then

  float rM = NEGINF, rS = 0.0f, pM = NEGINF, pS = 0.0f;
  float rv[PERT], pv[PERT];
  int   ri[PERT], pi[PERT];
#pragma unroll
  for (int j = 0; j < PERT; ++j) { rv[j] = NEGINF; pv[j] = NEGINF; ri[j] = -1; pi[j] = -1; }

  async_copy_b128(row + tid * 4, &buf[0][tid * 4]);
#pragma unroll 1
  for (int tt = 0; tt < NT; ++tt) {
    if (tt + 1 < NT) {
      async_copy_b128(row + (tt + 1) * TILE + tid * 4, &buf[(tt + 1) & 1][tid * 4]);
      wait_async_le1();
    } else {
      wait_async_0();
    }
    __syncthreads();
    const float* tb = buf[tt & 1];
    const int base = tt * TILE + tid * 4;
#pragma unroll
    for (int j = 0; j < 4; ++j) {
      float x = tb[tid * 4 + j];
      int   v = base + j;
      lse_update(rM, rS, x);                 // raw logsumexp (always)
      int   cnt = hlookup(hkey, hcnt, v);
      float p   = proc_val(x, cnt, rp, fp, pp, invt);
      if (needProcLse) lse_update(pM, pS, p); // uniform per-row branch
      topk_insert(rv, ri, x, v);
      topk_insert(pv, pi, p, v);
    }
    __syncthreads();
  }

  // block reduce the two (max, sum) pairs
  if (!needProcLse) { pM = NEGINF; pS = 0.0f; } // keep reduce NaN-free
  rm[tid] = rM; rs[tid] = rS; pm[tid] = pM; ps[tid] = pS;
  __syncthreads();
#pragma unroll 1
  for (int off = BLK / 2; off > 0; off >>= 1) {
    if (tid < off) {
      float mA = rm[tid], sA = rs[tid], mB = rm[tid + off], sB = rs[tid + off];
      float M  = fmaxf(mA, mB);
      rm[tid] = M;
      rs[tid] = sA * __expf(mA - M) + sB * __expf(mB - M);
      if (needProcLse) {
        mA = pm[tid]; sA = ps[tid]; mB = pm[tid + off]; sB = ps[tid + off];
        M = fmaxf(mA, mB);
        pm[tid] = M;
        ps[tid] = sA * __expf(mA - M) + sB * __expf(mB - M);
      }
    }
    __syncthreads();
  }

  float* wrow = ws + (size_t)b * ROWF;
  if (tid == 0) {
    wrow[0] = rm[0]; wrow[1] = rs[0];
    wrow[2] = needProcLse ? pm[0] : 0.0f;
    wrow[3] = needProcLse ? ps[0] : 1.0f;
  }
  float* rV = wrow + HDR;
  float* rI = rV + CAND;
  float* qV = rI + CAND;
  float* qI = qV + CAND;
#pragma unroll
  for (int j = 0; j < PERT; ++j) {
    rV[tid * PERT + j] = rv[j];
    rI[tid * PERT + j] = (float)ri[j];
    qV[tid * PERT + j] = pv[j];
    qI[tid * PERT + j] = (float)pi[j];
  }
}

// =====================================================================
// Kernel 2: candidate-space selection: top-20 raw logprobs + fused threshold
// =====================================================================
__global__ void __launch_bounds__(BLK) k2_select(
    const int* __restrict__ top_k, const float* __restrict__ top_p,
    float* __restrict__ ws, float* __restrict__ out) {
  __shared__ float cv[CAND];
  __shared__ int   ci[CAND];
  __shared__ float bv[BLK];
  __shared__ int   bi[BLK];
  __shared__ float sorted[NSORT];

  const int b   = blockIdx.x;
  const int tid = threadIdx.x;
  float* wrow = ws + (size_t)b * ROWF;
  const float Mraw = wrow[0], Sraw = wrow[1];
  const float Mproc = wrow[2], Sproc = wrow[3];
  const float lse = Mraw + __logf(Sraw);

  float* lpOut  = out + B + (size_t)b * (NLP + 1);
  float* idxOut = out + B + (size_t)B * (NLP + 1) + (size_t)b * (NLP + 1);

  // ---- pass 1: top-20 of raw candidates ----
  {
    const float* rV = wrow + HDR;
    const float* rI = rV + CAND;
#pragma unroll
    for (int j = 0; j < PERT; ++j) {
      cv[tid * PERT + j] = rV[tid * PERT + j];
      ci[tid * PERT + j] = (int)rI[tid * PERT + j];
    }
    __syncthreads();
#pragma unroll 1
    for (int r = 0; r < NLP; ++r) {
      float bestv = NEGINF; int bests = -1;
#pragma unroll
      for (int j = 0; j < PERT; ++j) {
        int s = tid * PERT + j;
        if (cv[s] > bestv) { bestv = cv[s]; bests = s; }
      }
      bv[tid] = bestv; bi[tid] = bests;
      __syncthreads();
#pragma unroll 1
      for (int off = BLK / 2; off > 0; off >>= 1) {
        if (tid < off) {
          if (bv[tid + off] > bv[tid]) { bv[tid] = bv[tid + off]; bi[tid] = bi[tid + off]; }
        }
        __syncthreads();
      }
      if (tid == 0) {
        int slot = bi[0];
        lpOut[1 + r]  = cv[slot] - lse;
        idxOut[1 + r] = (float)ci[slot];
        cv[slot] = NEGINF;
      }
      __syncthreads();
    }
  }

  // ---- pass 2: sorted top-64 of processed candidates ----
  {
    const float* qV = wrow + HDR + 2 * CAND;
#pragma unroll
    for (int j = 0; j < PERT; ++j) cv[tid * PERT + j] = qV[tid * PERT + j];
    __syncthreads();
#pragma unroll 1
    for (int r = 0; r < NSORT; ++r) {
      float bestv = NEGINF; int bests = -1;
#pragma unroll
      for (int j = 0; j < PERT; ++j) {
        int s = tid * PERT + j;
        if (cv[s] > bestv) { bestv = cv[s]; bests = s; }
      }
      bv[tid] = bestv; bi[tid] = bests;
      __syncthreads();
#pragma unroll 1
      for (int off = BLK / 2; off > 0; off >>= 1) {
        if (tid < off) {
          if (bv[tid + off] > bv[tid]) { bv[tid] = bv[tid + off]; bi[tid] = bi[tid + off]; }
        }
        __syncthreads();
      }
      if (tid == 0) {
        sorted[r] = cv[bi[0]];
        cv[bi[0]] = NEGINF;
      }
      __syncthreads();
    }
  }

  if (tid == 0) {
    const int   tk = top_k[b];
    const float tp = top_p[b];
    float thK = NEGINF;
    int   kk  = NSORT;
    if (tk > 0) {
      int ke = (tk < NSORT) ? tk : NSORT;
      thK = sorted[ke - 1];
      kk  = ke;
    }
    // softmax denominator over the kept set
    float Mn, D;
    if (tk > 0) {
      Mn = sorted[0]; D = 0.0f;
      for (int i = 0; i < kk; ++i) D += __expf(sorted[i] - Mn);
    } else {
      Mn = Mproc; D = Sproc;
    }
    // top-p: keep element while cumulative-before < top_p
    float cum = 0.0f, thP = NEGINF;
    for (int i = 0; i < kk; ++i) {
      if (cum >= tp) break;
      thP = sorted[i];
      cum += __expf(sorted[i] - Mn) / D;
    }
    if (tk <= 0 && cum < tp) thP = NEGINF; // tail outside candidates also kept
    wrow[4] = fmaxf(thK, thP);
  }
}

// =====================================================================
// Kernel 3: streamed pass -> gumbel argmax + greedy argmax + outputs
// =====================================================================
__global__ void __launch_bounds__(BLK) k3_sample(
    const float* __restrict__ logits, const float* __restrict__ temperature,
    const float* __restrict__ rep_pen, const float* __restrict__ freq_pen,
    const float* __restrict__ pres_pen, const int* __restrict__ out_ids,
    const float* __restrict__ ws, float* __restrict__ out) {
  __shared__ float buf[2][TILE];
  __shared__ int   hkey[HSZ];
  __shared__ int   hcnt[HSZ];
  __shared__ float sv[BLK];
  __shared__ int   si[BLK];
  __shared__ float sx[BLK];

  const int b   = blockIdx.x;
  const int tid = threadIdx.x;
  const float* row = logits + (size_t)b * V;

  build_hash(hkey, hcnt, out_ids + (size_t)b * L, tid);

  const float* wrow = ws + (size_t)b * ROWF;
  const float Mraw = wrow[0], Sraw = wrow[1];
  const float thr  = wrow[4];
  const float rp = rep_pen[b], fp = freq_pen[b], pp = pres_pen[b];
  float t = temperature[b];
  const bool greedy = (t < FEPS);
  t = greedy ? 1.0f : t;
  const float invt = 1.0f / t;

  float gBest = NEGINF; int gIdx = -1; float gRaw = 0.0f; // gumbel pick
  float pBest = NEGINF; int pIdx = -1; float pRaw = 0.0f; // greedy pick

  async_copy_b128(row + tid * 4, &buf[0][tid * 4]);
#pragma unroll 1
  for (int tt = 0; tt < NT; ++tt) {
    if (tt + 1 < NT) {
      async_copy_b128(row + (tt + 1) * TILE + tid * 4, &buf[(tt + 1) & 1][tid * 4]);
      wait_async_le1();
    } else {
      wait_async_0();
    }
    __syncthreads();
    const float* tb = buf[tt & 1];
    const int base = tt * TILE + tid * 4;
#pragma unroll
    for (int j = 0; j < 4; ++j) {
      float x = tb[tid * 4 + j];
      int   v = base + j;
      int   cnt = hlookup(hkey, hcnt, v);
      float p   = proc_val(x, cnt, rp, fp, pp, invt);
      if (p >= thr) {
        float u  = rng_uniform((uint64_t)b * (uint64_t)V + (uint64_t)v);
        float g  = -__logf(-__logf(u));
        float sc = p + g;
        if (sc > gBest) { gBest = sc; gIdx = v; gRaw = x; }
        if (p > pBest)  { pBest = p;  pIdx = v; pRaw = x; }
      }
    }
    __syncthreads();
  }

  // reduce gumbel argmax
  sv[tid] = gBest; si[tid] = gIdx; sx[tid] = gRaw;
  __syncthreads();
#pragma unroll 1
  for (int off = BLK / 2; off > 0; off >>= 1) {
    if (tid < off) {
      bool take = (sv[tid + off] > sv[tid]) ||
                  (sv[tid + off] == sv[tid] && si[tid + off] >= 0 &&
                   (si[tid] < 0 || si[tid + off] < si[tid]));
      if (take) { sv[tid] = sv[tid + off]; si[tid] = si[tid + off]; sx[tid] = sx[tid + off]; }
    }
    __syncthreads();
  }
  int   selG = si[0];
  float rawG = sx[0];
  __syncthreads();

  // reduce greedy argmax
  sv[tid] = pBest; si[tid] = pIdx; sx[tid] = pRaw;
  __syncthreads();
#pragma unroll 1
  for (int off = BLK / 2; off > 0; off >>= 1) {
    if (tid < off) {
      bool take = (sv[tid + off] > sv[tid]) ||
                  (sv[tid + off] == sv[tid] && si[tid + off] >= 0 &&
                   (si[tid] < 0 || si[tid + off] < si[tid]));
      if (take) { sv[tid] = sv[tid + off]; si[tid] = si[tid + off]; sx[tid] = sx[tid + off]; }
    }
    __syncthreads();
  }

  if (tid == 0) {
    int   sel  = greedy ? si[0] : selG;
    float rawx = greedy ? sx[0] : rawG;
    if (sel < 0) { sel = 0; rawx = row[0]; } // defensive; cannot normally happen
    const float lse = Mraw + __logf(Sraw);
    out[b] = (float)sel;                                         // sampled_token_ids
    out[B + (size_t)b * (NLP + 1)] = rawx - lse;                 // samp_lp (col 0)
    out[B + (size_t)B * (NLP + 1) + (size_t)b * (NLP + 1)] = (float)sel; // index col 0
  }
}

// =====================================================================
extern "C" void kernel_launch(void* const* d_in, const int* in_sizes, int n_in,
                              void* d_out, int out_size, void* d_ws, size_t ws_size,
                              hipStream_t stream) {
  const float* logits      = (const float*)d_in[0];
  const float* temperature = (const float*)d_in[1];
  const int*   top_k       = (const int*)d_in[2];
  const float* top_p       = (const float*)d_in[3];
  const float* rep_pen     = (const float*)d_in[4];
  const float* freq_pen    = (const float*)d_in[5];
  const float* pres_pen    = (const float*)d_in[6];
  const int*   out_ids     = (const int*)d_in[7];
  (void)in_sizes; (void)n_in; (void)out_size; (void)ws_size;

  float* out = (float*)d_out;
  float* ws  = (float*)d_ws;

  hipLaunchKernelGGL(k1_stats, dim3(B), dim3(BLK), 0, stream,
                     logits, temperature, top_k, rep_pen, freq_pen, pres_pen, out_ids, ws);
  hipLaunchKernelGGL(k2_select, dim3(B), dim3(BLK), 0, stream,
                     top_k, top_p, ws, out);
  hipLaunchKernelGGL(k3_sample, dim3(B), dim3(BLK), 0, stream,
                     logits, temperature, rep_pen, freq_pen, pres_pen, out_ids, ws, out);
}